// GraphConv_70231305224360
// MI455X (gfx1250) — compile-verified
//
#include <hip/hip_runtime.h>
#include <hip/hip_bf16.h>

typedef __attribute__((ext_vector_type(2))) float v2f;
typedef __attribute__((ext_vector_type(8))) float v8f;

#define GC_F 256   // input features (K of GEMM)
#define GC_C 256   // output channels

// ---------------------------------------------------------------------------
// Kernel 1: H = X @ W using V_WMMA_F32_16X16X4_F32 (exact fp32 tensor path).
// Block = 256 threads = 8 waves; block handles one 16-row tile of X.
// Each wave computes two adjacent 16x16 output tiles (col tiles 2w, 2w+1),
// reusing the A fragment across both WMMAs.
//
// A fragment (16x4 f32, ISA 7.12.2): lane L holds row M=L%16;
//   VGPR r holds K = 2*(L/16) + r.
// B fragment (4x16 f32, mirrored): lane L holds col N=L%16;
//   VGPR r holds K = 2*(L/16) + r.
// C/D (16x16 f32): lane L holds col N=L%16; VGPR v holds row M = v + 8*(L/16).
// ---------------------------------------------------------------------------
__global__ __launch_bounds__(256) void gc_gemm(const float* __restrict__ X,
                                               const float* __restrict__ W,
                                               float* __restrict__ H, int N) {
  const int lane    = threadIdx.x & 31;
  const int wave    = threadIdx.x >> 5;      // 0..7
  const int rowBase = blockIdx.x * 16;
  if (rowBase + 16 > N) return;              // block-uniform exit, EXEC stays full
  const int m     = lane & 15;
  const int khalf = lane >> 4;               // 0 or 1
  const int ct0   = (wave * 2) * 16;         // col tile 0 base
  const int ct1   = ct0 + 16;                // col tile 1 base

  const float* xrow = X + (size_t)(rowBase + m) * GC_F;
  v8f c0 = {};
  v8f c1 = {};
  for (int k0 = 0; k0 < GC_F; k0 += 4) {
    const int ka = k0 + 2 * khalf;
    v2f a, b0, b1;
    a.x = xrow[ka];
    a.y = xrow[ka + 1];
    const float* w0 = W + (size_t)ka * GC_C;   // row ka of W
    b0.x = w0[ct0 + m];
    b1.x = w0[ct1 + m];
    b0.y = w0[GC_C + ct0 + m];                 // row ka+1 of W
    b1.y = w0[GC_C + ct1 + m];
    c0 = __builtin_amdgcn_wmma_f32_16x16x4_f32(false, a, false, b0, (short)0, c0,
                                               false, false);
    c1 = __builtin_amdgcn_wmma_f32_16x16x4_f32(false, a, false, b1, (short)0, c1,
                                               false, false);
  }
#pragma unroll
  for (int v = 0; v < 8; ++v) {
    const int row = rowBase + v + 8 * khalf;
    H[(size_t)row * GC_C + ct0 + m] = c0[v];   // lanes 0-15: 64B contiguous run
    H[(size_t)row * GC_C + ct1 + m] = c1[v];
  }
}

// ---------------------------------------------------------------------------
// Kernel 2: edge-parallel scatter-add. One wave32 per edge.
// Each lane loads 8 contiguous channels of H[col] (2x float4, fully coalesced
// 1KB row gather -> L2-resident since H = 102MB < 192MB L2) and issues 8
// global_atomic_add_f32 into agg[row]. Row indices are ~uniform random over
// 100k nodes, so per-address contention is negligible and the RMW stream
// stays inside L2.
// ---------------------------------------------------------------------------
__global__ __launch_bounds__(256) void gc_scatter(const int* __restrict__ rows,
                                                  const int* __restrict__ cols,
                                                  const float* __restrict__ vals,
                                                  const float* __restrict__ H,
                                                  float* __restrict__ agg, int E) {
  const long long gtid = (long long)blockIdx.x * blockDim.x + threadIdx.x;
  const long long wid  = gtid >> 5;           // edge index (wave-uniform)
  if (wid >= E) return;
  const int lane = threadIdx.x & 31;

  const int   r = rows[wid];
  const int   c = cols[wid];
  const float v = vals[wid];

  const float4* hrow = (const float4*)(H + (size_t)c * GC_C);
  float*        arow = agg + (size_t)r * GC_C;

  const float4 h0 = hrow[lane * 2];
  const float4 h1 = hrow[lane * 2 + 1];
  const int base  = lane * 8;

  unsafeAtomicAdd(&arow[base + 0], v * h0.x);
  unsafeAtomicAdd(&arow[base + 1], v * h0.y);
  unsafeAtomicAdd(&arow[base + 2], v * h0.z);
  unsafeAtomicAdd(&arow[base + 3], v * h0.w);
  unsafeAtomicAdd(&arow[base + 4], v * h1.x);
  unsafeAtomicAdd(&arow[base + 5], v * h1.y);
  unsafeAtomicAdd(&arow[base + 6], v * h1.z);
  unsafeAtomicAdd(&arow[base + 7], v * h1.w);
}

// ---------------------------------------------------------------------------
// Kernel 3: out = relu(out + bias), float4-vectorized streaming pass.
// ---------------------------------------------------------------------------
__global__ __launch_bounds__(256) void gc_finalize(float* __restrict__ out,
                                                   const float* __restrict__ bias,
                                                   long long tot4) {
  const long long i = (long long)blockIdx.x * blockDim.x + threadIdx.x;
  if (i >= tot4) return;
  float4*       o4 = (float4*)out;
  const float4* b4 = (const float4*)bias;
  float4 vo = o4[i];
  float4 vb = b4[i & (GC_C / 4 - 1)];         // bias index = (elem % 256) / 4
  vo.x = fmaxf(vo.x + vb.x, 0.0f);
  vo.y = fmaxf(vo.y + vb.y, 0.0f);
  vo.z = fmaxf(vo.z + vb.z, 0.0f);
  vo.w = fmaxf(vo.w + vb.w, 0.0f);
  o4[i] = vo;
}

// ---------------------------------------------------------------------------
// Launch: zero agg (in d_out), GEMM into d_ws, scatter-add, bias+relu.
// ---------------------------------------------------------------------------
extern "C" void kernel_launch(void* const* d_in, const int* in_sizes, int n_in,
                              void* d_out, int out_size, void* d_ws, size_t ws_size,
                              hipStream_t stream) {
  const float* x    = (const float*)d_in[0];
  const int*   erow = (const int*)d_in[1];
  const int*   ecol = (const int*)d_in[2];
  const float* ev   = (const float*)d_in[3];
  const float* wk   = (const float*)d_in[4];
  const float* bias = (const float*)d_in[5];

  const int N = in_sizes[0] / GC_F;   // 100000
  const int E = in_sizes[1];          // 3200000

  float* H   = (float*)d_ws;          // N*C*4 = 102.4 MB scratch for X@W
  float* out = (float*)d_out;

  hipMemsetAsync(out, 0, (size_t)N * GC_C * sizeof(float), stream);

  gc_gemm<<<(N + 15) / 16, 256, 0, stream>>>(x, wk, H, N);

  const int scatterBlocks = (E + 7) / 8;      // 8 waves (edges) per block
  gc_scatter<<<scatterBlocks, 256, 0, stream>>>(erow, ecol, ev, H, out, E);

  const long long tot4 = (long long)N * GC_C / 4;
  gc_finalize<<<(int)((tot4 + 255) / 256), 256, 0, stream>>>(out, bias, tot4);
}